// GinNet_64991445123397
// MI455X (gfx1250) — compile-verified
//
#include <hip/hip_runtime.h>
#include <math.h>

#define N_NODES 100000
#define N_EDGES 3200000
#define N_FEAT 16
#define HID 64
#define N_GRAPHS 128
#define N_CLASSES 10

typedef __attribute__((ext_vector_type(2))) float v2f;
typedef __attribute__((ext_vector_type(8))) float v8f;

// ---------- init: agg = (1+eps) * x   (float4 vectorized) ----------
__global__ void init_h_kernel(const float4* __restrict__ xin,
                              const float* __restrict__ epsp,
                              float4* __restrict__ out, long long total4) {
    long long tid = (long long)blockIdx.x * blockDim.x + threadIdx.x;
    if (tid >= total4) return;
    float s = 1.0f + *epsp;
    float4 v = xin[tid];
    float4 r;
    r.x = s * v.x; r.y = s * v.y; r.z = s * v.z; r.w = s * v.w;
    out[tid] = r;
}

// ---------- scatter-add: agg[dst] += x[src], one thread per (edge, float4 chunk) ----------
template <int F>
__global__ void scatter_add_kernel(const float* __restrict__ xin,
                                   const int* __restrict__ src,
                                   const int* __restrict__ dst,
                                   float* __restrict__ agg, long long total4) {
    long long tid = (long long)blockIdx.x * blockDim.x + threadIdx.x;
    if (tid >= total4) return;
    constexpr int C = F / 4;                 // float4 chunks per row (power of two)
    int e = (int)(tid / C);
    int c = (int)(tid % C);
    int s = src[e];
    int d = dst[e];
    float4 v = *((const float4*)(xin + (size_t)s * F) + c);   // global_load_b128 gather
    float* ar = agg + (size_t)d * F + c * 4;
    atomicAdd(ar + 0, v.x);
    atomicAdd(ar + 1, v.y);
    atomicAdd(ar + 2, v.z);
    atomicAdd(ar + 3, v.w);
}

// ---------- fused MLP: tanh(hin @ W + b) via V_WMMA_F32_16X16X4_F32 ----------
// hin: [N_NODES, K]; W: [K, HID] staged in LDS; b: [HID].
// FUSE_POOL=false: write hout [N_NODES, HID].
// FUSE_POOL=true : atomicAdd tanh result into pooled[batch[row]] (graph pooling fused).
template <int K, bool FUSE_POOL>
__global__ void mlp_wmma_kernel(const float* __restrict__ hin,
                                const float* __restrict__ W,
                                const float* __restrict__ b,
                                float* __restrict__ hout,
                                const int* __restrict__ batch,
                                float* __restrict__ pooled, int nTiles) {
    __shared__ float Wlds[K * HID];
    // cooperative stage of W into LDS (K*HID*4 bytes: 4KB or 16KB)
    constexpr int NV4 = K * HID / 4;
    for (int i = threadIdx.x; i < NV4; i += blockDim.x)
        ((float4*)Wlds)[i] = ((const float4*)W)[i];
    __syncthreads();

    int wave = (int)((blockIdx.x * blockDim.x + threadIdx.x) >> 5);
    int lane = threadIdx.x & 31;
    if (wave >= nTiles) return;          // wave-uniform guard AFTER barrier: EXEC all-1s

    const int row0 = wave * 16;
    const int m    = lane & 15;          // N column within tile / M row for A
    const int half = lane >> 4;          // selects K sub-pair

    v8f acc[4];
#pragma unroll
    for (int t = 0; t < 4; ++t) {
        float bv = b[t * 16 + m];        // bias depends only on output column N
#pragma unroll
        for (int v = 0; v < 8; ++v) acc[t][v] = bv;
    }

    for (int k = 0; k < K; k += 4) {
        // A fragment (16x4): lane holds hin[row0+m][k+2*half .. +1] (contiguous b64)
        const float* ap = hin + (size_t)(row0 + m) * K + (k + 2 * half);
        v2f a;
        a.x = ap[0];
        a.y = ap[1];
#pragma unroll
        for (int t = 0; t < 4; ++t) {
            // B fragment (4x16) from LDS: VGPR0 = K rows {0|2}, VGPR1 = {1|3}, N = m
            int ncol = t * 16 + m;
            v2f bb;
            bb.x = Wlds[(k + 2 * half + 0) * HID + ncol];
            bb.y = Wlds[(k + 2 * half + 1) * HID + ncol];
            acc[t] = __builtin_amdgcn_wmma_f32_16x16x4_f32(
                false, a, false, bb, (short)0, acc[t], false, false);
        }
    }

    // C/D layout: VGPR v, lane -> M = v + 8*half, N = t*16 + m
    if constexpr (FUSE_POOL) {
        int gv[8];
#pragma unroll
        for (int v = 0; v < 8; ++v) gv[v] = batch[row0 + v + 8 * half];
#pragma unroll
        for (int t = 0; t < 4; ++t) {
#pragma unroll
            for (int v = 0; v < 8; ++v) {
                int N = t * 16 + m;
                atomicAdd(&pooled[(size_t)gv[v] * HID + N], tanhf(acc[t][v]));
            }
        }
    } else {
#pragma unroll
        for (int t = 0; t < 4; ++t) {
#pragma unroll
            for (int v = 0; v < 8; ++v) {
                int M = v + 8 * half;
                int N = t * 16 + m;
                hout[(size_t)(row0 + M) * HID + N] = tanhf(acc[t][v]);
            }
        }
    }
}

// ---------- zero small buffer ----------
__global__ void zero_kernel(float* __restrict__ p, int n) {
    int tid = blockIdx.x * blockDim.x + threadIdx.x;
    if (tid < n) p[tid] = 0.0f;
}

// ---------- final FC: out = tanh(pooled @ fcW + fcb), [128,64]x[64,10] ----------
__global__ void fc_kernel(const float* __restrict__ pooled,
                          const float* __restrict__ Wfc,
                          const float* __restrict__ bfc,
                          float* __restrict__ out) {
    int tid = blockIdx.x * blockDim.x + threadIdx.x;
    if (tid >= N_GRAPHS * N_CLASSES) return;
    int g = tid / N_CLASSES;
    int c = tid % N_CLASSES;
    float s = bfc[c];
#pragma unroll
    for (int k = 0; k < HID; ++k) s += pooled[g * HID + k] * Wfc[k * N_CLASSES + c];
    out[tid] = tanhf(s);
}

extern "C" void kernel_launch(void* const* d_in, const int* in_sizes, int n_in,
                              void* d_out, int out_size, void* d_ws, size_t ws_size,
                              hipStream_t stream) {
    // setup_inputs order:
    // 0:x 1:edge_index 2:batch 3:eps1 4:W1 5:b1 6:eps2 7:W2 8:b2 9:eps3 10:W3 11:b3 12:fcW 13:fcb
    const float* x     = (const float*)d_in[0];
    const int*   eidx  = (const int*)d_in[1];
    const int*   batch = (const int*)d_in[2];
    const float* eps1  = (const float*)d_in[3];
    const float* W1    = (const float*)d_in[4];
    const float* b1    = (const float*)d_in[5];
    const float* eps2  = (const float*)d_in[6];
    const float* W2    = (const float*)d_in[7];
    const float* b2    = (const float*)d_in[8];
    const float* eps3  = (const float*)d_in[9];
    const float* W3    = (const float*)d_in[10];
    const float* b3    = (const float*)d_in[11];
    const float* fcW   = (const float*)d_in[12];
    const float* fcb   = (const float*)d_in[13];

    const int* src = eidx;             // edge_index[0, :]
    const int* dst = eidx + N_EDGES;   // edge_index[1, :]

    float* A      = (float*)d_ws;                 // [N_NODES, HID] scratch (pre-MLP)
    float* B      = A + (size_t)N_NODES * HID;    // [N_NODES, HID] scratch (post-MLP)
    float* pooled = B + (size_t)N_NODES * HID;    // [N_GRAPHS, HID]

    const int TPB = 256;
    const int nTiles    = N_NODES / 16;                   // 6250 exactly
    const int mlpBlocks = (nTiles * 32 + TPB - 1) / TPB;  // 8 waves / block

    auto blocks = [](long long total) { return (int)((total + 255) / 256); };

    // ---- layer 1 (K = N_FEAT = 16) ----
    {
        long long tI4 = (long long)N_NODES * N_FEAT / 4;
        long long tE4 = (long long)N_EDGES * (N_FEAT / 4);
        init_h_kernel<<<blocks(tI4), TPB, 0, stream>>>((const float4*)x, eps1, (float4*)A, tI4);
        scatter_add_kernel<N_FEAT><<<blocks(tE4), TPB, 0, stream>>>(x, src, dst, A, tE4);
        mlp_wmma_kernel<N_FEAT, false><<<mlpBlocks, TPB, 0, stream>>>(A, W1, b1, B,
                                                                      batch, pooled, nTiles);
    }
    // ---- layer 2 (K = HID = 64) ----
    {
        long long tI4 = (long long)N_NODES * HID / 4;
        long long tE4 = (long long)N_EDGES * (HID / 4);
        init_h_kernel<<<blocks(tI4), TPB, 0, stream>>>((const float4*)B, eps2, (float4*)A, tI4);
        scatter_add_kernel<HID><<<blocks(tE4), TPB, 0, stream>>>(B, src, dst, A, tE4);
        mlp_wmma_kernel<HID, false><<<mlpBlocks, TPB, 0, stream>>>(A, W2, b2, B,
                                                                   batch, pooled, nTiles);
    }
    // ---- layer 3 (fused graph pooling in the MLP epilogue) ----
    {
        long long tI4 = (long long)N_NODES * HID / 4;
        long long tE4 = (long long)N_EDGES * (HID / 4);
        init_h_kernel<<<blocks(tI4), TPB, 0, stream>>>((const float4*)B, eps3, (float4*)A, tI4);
        scatter_add_kernel<HID><<<blocks(tE4), TPB, 0, stream>>>(B, src, dst, A, tE4);
        zero_kernel<<<blocks(N_GRAPHS * HID), TPB, 0, stream>>>(pooled, N_GRAPHS * HID);
        mlp_wmma_kernel<HID, true><<<mlpBlocks, TPB, 0, stream>>>(A, W3, b3, B,
                                                                  batch, pooled, nTiles);
    }
    // ---- FC head ----
    fc_kernel<<<blocks(N_GRAPHS * N_CLASSES), TPB, 0, stream>>>(pooled, fcW, fcb,
                                                                (float*)d_out);
}